// MyLoss_82102594830976
// MI455X (gfx1250) — compile-verified
//
#include <hip/hip_runtime.h>

// Problem shape (from reference):
//   outputs: (8, 527, 50257) f32, targets: (8, 512) int, ratings: (8,) int, stage: scalar int
//   shift = MAX_TOPIC_LEN + SEQ_LEN = 15, shifted seq len = 527 - 15 - 1 = 511
#define NB     8
#define TFULL  527
#define NV     50257
#define NL     512
#define SHIFT  15
#define TS     511
#define NROWS  (NB * TS)     // 4088 rows of logsumexp
#define NEAR0  1e-10f
#define BLK    256           // 8 wave32s per row

typedef __attribute__((ext_vector_type(2))) float v2f;
typedef __attribute__((ext_vector_type(8))) float v8f;

// One workgroup per (b, t) row: streaming online logsumexp over NV f32 logits.
// HBM-bound: 822 MB total -> ~35 us floor at 23.3 TB/s.
__global__ __launch_bounds__(BLK)
void row_nll_kernel(const float* __restrict__ outs,
                    const int*   __restrict__ targets,
                    float*       __restrict__ nll)
{
    __shared__ float wred[BLK / 32];   // per-wave maxima
    __shared__ float tvals[BLK];       // rescaled per-thread partial sums

    const int row = blockIdx.x;
    const int b   = row / TS;
    const int t   = row - b * TS;
    const float* __restrict__ rowp =
        outs + (size_t)(b * TFULL + SHIFT + t) * (size_t)NV;

    const int tid  = threadIdx.x;
    const int lane = tid & 31;
    const int wid  = tid >> 5;

    float m = -__builtin_inff();
    float s = 0.0f;

    // Bulk: b128 streaming loads, chunk-of-4 online logsumexp (~1.25 exp/elem).
    const float4* __restrict__ rp4 = (const float4*)rowp;
    const int n4 = NV >> 2;                       // 12564 float4s
    for (int i = tid; i < n4; i += BLK) {
        __builtin_prefetch(rp4 + i + 8 * BLK, 0, 0);   // global_prefetch_b8, ~32KB ahead
        float4 x = rp4[i];
        float cm = fmaxf(fmaxf(x.x, x.y), fmaxf(x.z, x.w));
        float M  = fmaxf(m, cm);
        s = s * __expf(m - M)
          + __expf(x.x - M) + __expf(x.y - M)
          + __expf(x.z - M) + __expf(x.w - M);
        m = M;
    }
    // Tail (NV % 4 == 1 -> exactly one element, handled by tid 0).
    for (int i = (n4 << 2) + tid; i < NV; i += BLK) {
        float x = rowp[i];
        float M = fmaxf(m, x);
        s = s * __expf(m - M) + __expf(x - M);
        m = M;
    }

    // ---- block max (wave32 shuffle tree + LDS across 8 waves) ----
    float wm = m;
    for (int off = 16; off; off >>= 1)
        wm = fmaxf(wm, __shfl_xor(wm, off));
    if (lane == 0) wred[wid] = wm;
    __syncthreads();
    float Mb = wred[0];
    #pragma unroll
    for (int i = 1; i < BLK / 32; ++i) Mb = fmaxf(Mb, wred[i]);

    // Rescale partial sums to the common max; stash for the block-wide sum.
    tvals[tid] = s * __expf(m - Mb);
    __syncthreads();

    if (tid < 32) {            // wave 0 fully active -> EXEC all-ones for WMMA
        float S;
#if __has_builtin(__builtin_amdgcn_wmma_f32_16x16x4_f32)
        // Sum 256 f32 values with 4x V_WMMA_F32_16X16X4_F32 against a ones
        // B-matrix, accumulating in C.  A 16x4 layout (ISA 7.12.2): lanes 0-15
        // hold K=0,1 in VGPR0,1; lanes 16-31 hold K=2,3.  D[m,n] is replicated
        // across n, so lanes<16 carry rows 0-7, lanes>=16 carry rows 8-15.
        v8f c = {};
        v2f ones; ones.x = 1.0f; ones.y = 1.0f;
        #pragma unroll
        for (int i = 0; i < 4; ++i) {
            v2f a;
            a.x = tvals[64 * i + tid];
            a.y = tvals[64 * i + 32 + tid];
            c = __builtin_amdgcn_wmma_f32_16x16x4_f32(
                    false, a, false, ones, (short)0, c, false, false);
        }
        float r = c[0] + c[1] + c[2] + c[3] + c[4] + c[5] + c[6] + c[7];
        S = r + __shfl_xor(r, 16);           // rows 0-7 + rows 8-15
#else
        float r = 0.0f;
        for (int i = lane; i < BLK; i += 32) r += tvals[i];
        for (int off = 16; off; off >>= 1) r += __shfl_xor(r, off);
        S = r;
#endif
        if (tid == 0) {
            const int label = targets[b * NL + (t + 1)];
            const float lg  = rowp[label];
            nll[row] = (Mb + logf(S)) - lg;  // logsumexp - logit[label]
        }
    }
}

// Tiny epilogue: per-batch mean NLL -> BCE on p = exp(-ce) -> mean over batch.
// One wave32 per batch element.
__global__ __launch_bounds__(BLK)
void finalize_kernel(const float* __restrict__ nll,
                     const int*   __restrict__ ratings,
                     const int*   __restrict__ stage,
                     float*       __restrict__ out)
{
    __shared__ float lossb[NB];
    const int tid  = threadIdx.x;
    const int w    = tid >> 5;        // wave id == batch index
    const int lane = tid & 31;

    float acc = 0.0f;
    for (int i = lane; i < TS; i += 32) acc += nll[w * TS + i];
    for (int off = 16; off; off >>= 1) acc += __shfl_xor(acc, off);

    if (lane == 0) {
        const float ce = acc / (float)TS;
        const float p  = expf(-ce);
        const int thresh = (stage[0] == 1) ? 4 : 3;
        const float y  = (ratings[w] > thresh) ? 1.0f : 0.0f;
        lossb[w] = -y * logf(p + NEAR0) - (1.0f - y) * logf(1.0f - p + NEAR0);
    }
    __syncthreads();
    if (tid == 0) {
        float r = 0.0f;
        #pragma unroll
        for (int i = 0; i < NB; ++i) r += lossb[i];
        out[0] = r * (1.0f / (float)NB);
    }
}

extern "C" void kernel_launch(void* const* d_in, const int* in_sizes, int n_in,
                              void* d_out, int out_size, void* d_ws, size_t ws_size,
                              hipStream_t stream)
{
    const float* outputs = (const float*)d_in[0];
    const int*   targets = (const int*)  d_in[1];
    const int*   ratings = (const int*)  d_in[2];
    const int*   stage   = (const int*)  d_in[3];

    float* nll = (float*)d_ws;                    // NROWS floats of scratch

    row_nll_kernel<<<NROWS, BLK, 0, stream>>>(outputs, targets, nll);
    finalize_kernel<<<1, BLK, 0, stream>>>(nll, ratings, stage, (float*)d_out);
}